// GCN_70875550319061
// MI455X (gfx1250) — compile-verified
//
#include <hip/hip_runtime.h>
#include <math.h>

typedef __attribute__((ext_vector_type(2))) float v2f;
typedef __attribute__((ext_vector_type(8))) float v8f;

#define NN 100000
#define EE 1600000
#define QQ 1000000

// ---------------- degree histogram (float atomics are exact for integer counts) ----------
__global__ void GCN_deg_kernel(const int* __restrict__ esrc, const int* __restrict__ edst,
                               float* __restrict__ dout, float* __restrict__ din) {
    int e = blockIdx.x * blockDim.x + threadIdx.x;
    if (e >= EE) return;
    unsafeAtomicAdd(&dout[esrc[e]], 1.0f);
    unsafeAtomicAdd(&din[edst[e]], 1.0f);
}

// d holds [out_deg | in_deg] contiguously: 2N entries -> rsqrt(max(x,1))
__global__ void GCN_norm_kernel(float* __restrict__ d) {
    int i = blockIdx.x * blockDim.x + threadIdx.x;
    if (i >= 2 * NN) return;
    d[i] = rsqrtf(fmaxf(d[i], 1.0f));
}

// ---------------- layer-1 aggregation: agg1[dst] += feat[src] * out_norm[src] ------------
// one wave32 per edge, float4 (global_load_b128) per lane, 4 fp32 atomics per lane
__global__ void GCN_scatter1_kernel(const float* __restrict__ feat,
                                    const int* __restrict__ esrc, const int* __restrict__ edst,
                                    const float* __restrict__ onorm, float* __restrict__ agg) {
    int t = blockIdx.x * blockDim.x + threadIdx.x;
    int e = t >> 5, lane = t & 31;
    int s = esrc[e], d = edst[e];
    float sc = onorm[s];
    const float4 v = *(const float4*)(feat + (size_t)s * 128 + lane * 4);
    float* dp = agg + (size_t)d * 128 + lane * 4;
    unsafeAtomicAdd(dp + 0, v.x * sc);
    unsafeAtomicAdd(dp + 1, v.y * sc);
    unsafeAtomicAdd(dp + 2, v.z * sc);
    unsafeAtomicAdd(dp + 3, v.w * sc);
}

// ---------------- GEMM1: H = relu(agg1 @ W1 * in_norm[:,None] + b1) ---------------------
// fp32 WMMA 16x16x4; block=256 (8 waves) = 8 col tiles of 16; grid = N/16 row tiles
__global__ void __launch_bounds__(256)
GCN_gemm1_kernel(const float* __restrict__ A, const float* __restrict__ W,
                 const float* __restrict__ bias, const float* __restrict__ inorm,
                 float* __restrict__ H) {
    __shared__ float lW[128 * 128];                 // 64 KB of 320 KB WGP LDS
    int tid = threadIdx.x;
    {
        const float4* Wv = (const float4*)W;
        float4* lWv = (float4*)lW;
#pragma unroll
        for (int i = 0; i < 16; ++i) lWv[tid + 256 * i] = Wv[tid + 256 * i];
    }
    __syncthreads();

    int wave = tid >> 5, lane = tid & 31;
    int row0 = blockIdx.x * 16, col0 = wave * 16;
    int r = lane & 15, kh = lane >> 4;              // kh: lane half selects K pair / M half

    v8f c = {};
    const float* Ap = A + (size_t)(row0 + r) * 128 + 2 * kh;   // A frag: row r, K={0,1}|{2,3}
    const float* Bp = lW + (size_t)(2 * kh) * 128 + col0 + r;  // B frag: col r, same K pair
#pragma unroll
    for (int kt = 0; kt < 32; ++kt) {               // K = 128 in steps of 4
        v2f a = *(const v2f*)(Ap + kt * 4);
        v2f b;
        b.x = Bp[kt * 512];                         // k = kt*4 + 2*kh
        b.y = Bp[kt * 512 + 128];                   // k+1
        c = __builtin_amdgcn_wmma_f32_16x16x4_f32(false, a, false, b, (short)0, c, false, false);
    }
    int n = col0 + r;
    float bb = bias[n];
#pragma unroll
    for (int v = 0; v < 8; ++v) {                   // C layout: VGPR v -> M = v + 8*kh
        int m = row0 + v + 8 * kh;
        float val = fmaf(c[v], inorm[m], bb);
        H[(size_t)m * 128 + n] = fmaxf(val, 0.0f);
    }
}

// ---------------- GEMM2: T = (H * out_norm[:,None]) @ W2 (128 -> 64) --------------------
__global__ void __launch_bounds__(128)
GCN_gemm2_kernel(const float* __restrict__ Hin, const float* __restrict__ W,
                 const float* __restrict__ onorm, float* __restrict__ T) {
    __shared__ float lW[128 * 64];                  // 32 KB
    int tid = threadIdx.x;
    {
        const float4* Wv = (const float4*)W;
        float4* lWv = (float4*)lW;
#pragma unroll
        for (int i = 0; i < 16; ++i) lWv[tid + 128 * i] = Wv[tid + 128 * i];
    }
    __syncthreads();

    int wave = tid >> 5, lane = tid & 31;
    int row0 = blockIdx.x * 16, col0 = wave * 16;
    int r = lane & 15, kh = lane >> 4;
    float sc = onorm[row0 + r];                     // scale A rows on load

    v8f c = {};
    const float* Ap = Hin + (size_t)(row0 + r) * 128 + 2 * kh;
    const float* Bp = lW + (size_t)(2 * kh) * 64 + col0 + r;
#pragma unroll
    for (int kt = 0; kt < 32; ++kt) {
        v2f a = *(const v2f*)(Ap + kt * 4);
        a.x *= sc; a.y *= sc;
        v2f b;
        b.x = Bp[kt * 256];
        b.y = Bp[kt * 256 + 64];
        c = __builtin_amdgcn_wmma_f32_16x16x4_f32(false, a, false, b, (short)0, c, false, false);
    }
    int n = col0 + r;
#pragma unroll
    for (int v = 0; v < 8; ++v) {
        int m = row0 + v + 8 * kh;
        T[(size_t)m * 64 + n] = c[v];
    }
}

// ---------------- layer-2 aggregation: h2[dst] += T[src] --------------------------------
__global__ void GCN_scatter2_kernel(const float* __restrict__ T,
                                    const int* __restrict__ esrc, const int* __restrict__ edst,
                                    float* __restrict__ H2) {
    int t = blockIdx.x * blockDim.x + threadIdx.x;
    int e = t >> 5, lane = t & 31;
    int s = esrc[e], d = edst[e];
    const float2 v = *(const float2*)(T + (size_t)s * 64 + lane * 2);
    float* dp = H2 + (size_t)d * 64 + lane * 2;
    unsafeAtomicAdd(dp + 0, v.x);
    unsafeAtomicAdd(dp + 1, v.y);
}

// ---------------- h2 = h2 * in_norm[:,None] + b2 (no relu on layer 2) -------------------
__global__ void GCN_finalize_kernel(float* __restrict__ H2, const float* __restrict__ inorm,
                                    const float* __restrict__ b2) {
    int i = blockIdx.x * blockDim.x + threadIdx.x;  // over N*16 float4s
    if (i >= NN * 16) return;
    int row = i >> 4;
    int c4 = (i & 15) * 4;
    float s = inorm[row];
    float4 v = ((float4*)H2)[i];
    v.x = fmaf(v.x, s, b2[c4 + 0]);
    v.y = fmaf(v.y, s, b2[c4 + 1]);
    v.z = fmaf(v.z, s, b2[c4 + 2]);
    v.w = fmaf(v.w, s, b2[c4 + 3]);
    ((float4*)H2)[i] = v;
}

// ---------------- edge scoring: sigmoid(dot(h2[src], h2[dst])) --------------------------
// one wave32 per query; float2 per lane covers 64 dims; shfl-xor tree reduce
__global__ void GCN_score_kernel(const float* __restrict__ H2, const int* __restrict__ qs,
                                 const int* __restrict__ qd, float* __restrict__ out) {
    int t = blockIdx.x * blockDim.x + threadIdx.x;
    int q = t >> 5, lane = t & 31;
    int s = qs[q], d = qd[q];
    float2 a = *(const float2*)(H2 + (size_t)s * 64 + lane * 2);
    float2 b = *(const float2*)(H2 + (size_t)d * 64 + lane * 2);
    float p = a.x * b.x + a.y * b.y;
#pragma unroll
    for (int off = 16; off > 0; off >>= 1) p += __shfl_xor(p, off, 32);
    if (lane == 0) out[q] = 1.0f / (1.0f + __expf(-p));
}

extern "C" void kernel_launch(void* const* d_in, const int* in_sizes, int n_in,
                              void* d_out, int out_size, void* d_ws, size_t ws_size,
                              hipStream_t stream) {
    const float* feat = (const float*)d_in[0];
    const int* esrc   = (const int*)d_in[1];
    const int* edst   = (const int*)d_in[2];
    const int* qsrc   = (const int*)d_in[3];
    const int* qdst   = (const int*)d_in[4];
    const float* W1   = (const float*)d_in[5];
    const float* b1   = (const float*)d_in[6];
    const float* W2   = (const float*)d_in[7];
    const float* b2   = (const float*)d_in[8];
    float* out = (float*)d_out;

    float* ws    = (float*)d_ws;
    float* onorm = ws;                              // N
    float* inorm = ws + NN;                         // N (contiguous after onorm)
    float* agg1  = ws + 2 * NN;                     // N*128
    float* h1    = agg1 + (size_t)NN * 128;         // N*128
    float* t2    = h1 + (size_t)NN * 128;           // N*64
    float* h2    = t2 + (size_t)NN * 64;            // N*64

    // zero degrees + agg1 (contiguous), and h2
    hipMemsetAsync(ws, 0, (size_t)(2 * NN + (size_t)NN * 128) * sizeof(float), stream);
    hipMemsetAsync(h2, 0, (size_t)NN * 64 * sizeof(float), stream);

    GCN_deg_kernel<<<(EE + 255) / 256, 256, 0, stream>>>(esrc, edst, onorm, inorm);
    GCN_norm_kernel<<<(2 * NN + 255) / 256, 256, 0, stream>>>(onorm);
    GCN_scatter1_kernel<<<EE / 8, 256, 0, stream>>>(feat, esrc, edst, onorm, agg1);
    GCN_gemm1_kernel<<<NN / 16, 256, 0, stream>>>(agg1, W1, b1, inorm, h1);
    GCN_gemm2_kernel<<<NN / 16, 128, 0, stream>>>(h1, W2, onorm, t2);
    GCN_scatter2_kernel<<<EE / 8, 256, 0, stream>>>(t2, esrc, edst, h2);
    GCN_finalize_kernel<<<(NN * 16 + 255) / 256, 256, 0, stream>>>(h2, inorm, b2);
    GCN_score_kernel<<<QQ / 8, 256, 0, stream>>>(h2, qsrc, qdst, out);
}